// AdaptiveScalingFilters_80582176407632
// MI455X (gfx1250) — compile-verified
//
#include <hip/hip_runtime.h>
#include <hip/hip_bf16.h>
#include <math.h>

// ---------------------------------------------------------------------------
// Morlet filter bank builder for MI455X (gfx1250).
//   resize (antialiased linear)  ->  2x f32 WMMA GEMMs per scale
//   rotate + fftshift + m*exp(ip) -> fused gather/store kernel
// Output-BW bound: 179MB @ 23.3TB/s ~= 7.7us floor; GEMMs (~15 GFLOP f32 via
// V_WMMA_F32_16X16X4_F32) and 45M sin/cos are negligible next to that.
// ---------------------------------------------------------------------------

typedef __attribute__((ext_vector_type(2))) float v2f;
typedef __attribute__((ext_vector_type(8))) float v8f;

#define NSRC 512   // primary filter size (input images are 512x512)

// ---------------------------------------------------------------------------
// Stage 0: weight matrix W[k, n] (row-major [512, s]) reproducing
// jax.image.resize(method="linear", antialias=True):
//   inv_scale = 512/s ; kernel_scale = max(inv_scale, 1)
//   sample_f(n) = (n+0.5)*inv_scale - 0.5
//   w = tri(|sample_f - k| / kernel_scale), column-normalized.
// ---------------------------------------------------------------------------
__global__ void weights_kernel(float* __restrict__ W, int s) {
    int idx = blockIdx.x * blockDim.x + threadIdx.x;
    if (idx >= NSRC * s) return;
    int k = idx / s;
    int n = idx - k * s;

    float inv_scale = (float)NSRC / (float)s;
    float ks = inv_scale > 1.0f ? inv_scale : 1.0f;
    float sf = ((float)n + 0.5f) * inv_scale - 0.5f;

    float w = fmaxf(1.0f - fabsf(sf - (float)k) / ks, 0.0f);

    // column sum over the (small) support for normalization
    int k0 = (int)ceilf(sf - ks);  if (k0 < 0) k0 = 0;
    int k1 = (int)floorf(sf + ks); if (k1 > NSRC - 1) k1 = NSRC - 1;
    float total = 0.0f;
    for (int kk = k0; kk <= k1; ++kk)
        total += fmaxf(1.0f - fabsf(sf - (float)kk) / ks, 0.0f);

    W[idx] = w / total;
}

// ---------------------------------------------------------------------------
// f32 WMMA tile GEMMs: one wave per 16x16 output tile, K=512 in steps of 4.
// A-matrix (16x4 f32) lane layout per ISA: lane<16 -> K={0,1}, lane>=16 -> K={2,3}.
// C/D layout: lane holds D[v + 8*(lane>>4), lane&15] in VGPR v.
// ---------------------------------------------------------------------------

// GEMM1:  T[n, c] = sum_k W[k, n] * M[k, c]      T: [s, 512]
__global__ void gemm1_kernel(const float* __restrict__ M0,
                             const float* __restrict__ M1,
                             const float* __restrict__ W,
                             float* __restrict__ T, int s) {
    int wavesPerBlock = blockDim.x >> 5;
    int wave = blockIdx.x * wavesPerBlock + (threadIdx.x >> 5);
    int lane = threadIdx.x & 31;
    int tilesN = NSRC / 16;
    int totalTiles = (s / 16) * tilesN;
    if (wave >= totalTiles) return;               // whole-wave exit (EXEC stays all-1s)

    int tm = (wave / tilesN) * 16;                // output row  (n  dim, size s)
    int tn = (wave % tilesN) * 16;                // output col  (c  dim, size 512)
    const float* Msrc = (blockIdx.z == 0) ? M0 : M1;
    float* Tdst = T + (size_t)blockIdx.z * (size_t)s * NSRC;

    int l16  = lane & 15;
    int koff = (lane < 16) ? 0 : 2;

    v8f acc = {};
    for (int k = 0; k < NSRC; k += 4) {
        v2f a, b;
        // A[m, kk] = W[kk, tm + m]   (transposed read of W)
        a.x = W[(size_t)(k + koff)     * s + tm + l16];
        a.y = W[(size_t)(k + koff + 1) * s + tm + l16];
        // B[kk, c] = M[kk, tn + c]
        b.x = Msrc[(size_t)(k + koff)     * NSRC + tn + l16];
        b.y = Msrc[(size_t)(k + koff + 1) * NSRC + tn + l16];
        acc = __builtin_amdgcn_wmma_f32_16x16x4_f32(
                  false, a, false, b, (short)0, acc, false, false);
    }
    int rbase = tm + ((lane >> 4) << 3);
    #pragma unroll
    for (int v = 0; v < 8; ++v)
        Tdst[(size_t)(rbase + v) * NSRC + tn + l16] = acc[v];
}

// GEMM2:  R[n, m] = sum_c T[n, c] * W[c, m]      R: [s, s]
__global__ void gemm2_kernel(const float* __restrict__ T,
                             const float* __restrict__ W,
                             float* __restrict__ R, int s) {
    int wavesPerBlock = blockDim.x >> 5;
    int wave = blockIdx.x * wavesPerBlock + (threadIdx.x >> 5);
    int lane = threadIdx.x & 31;
    int tilesN = s / 16;
    int totalTiles = tilesN * tilesN;
    if (wave >= totalTiles) return;

    int tm = (wave / tilesN) * 16;
    int tn = (wave % tilesN) * 16;
    const float* Tsrc = T + (size_t)blockIdx.z * (size_t)s * NSRC;
    float* Rdst = R + (size_t)blockIdx.z * (size_t)s * s;

    int l16  = lane & 15;
    int koff = (lane < 16) ? 0 : 2;

    v8f acc = {};
    for (int k = 0; k < NSRC; k += 4) {
        v2f a, b;
        // A[n, kk] = T[tm + n, kk]
        a.x = Tsrc[(size_t)(tm + l16) * NSRC + k + koff];
        a.y = Tsrc[(size_t)(tm + l16) * NSRC + k + koff + 1];
        // B[kk, m] = W[kk, tn + m]
        b.x = W[(size_t)(k + koff)     * s + tn + l16];
        b.y = W[(size_t)(k + koff + 1) * s + tn + l16];
        acc = __builtin_amdgcn_wmma_f32_16x16x4_f32(
                  false, a, false, b, (short)0, acc, false, false);
    }
    int rbase = tm + ((lane >> 4) << 3);
    #pragma unroll
    for (int v = 0; v < 8; ++v)
        Rdst[(size_t)(rbase + v) * s + tn + l16] = acc[v];
}

// ---------------------------------------------------------------------------
// Stage 2: for each rotation l and output pixel (y,x):
//   (yy,xx) = inverse fftshift coords ; bilinear-rotate (zero pad) both the
//   resized mag and phase images ; write m*cos(p), m*sin(p) interleaved.
// Rm = R, Rp = R + s*s.
// ---------------------------------------------------------------------------
__global__ void finalize_kernel(const float* __restrict__ R,
                                float* __restrict__ out, int s,
                                unsigned long long outBaseComplex) {
    size_t idx = (size_t)blockIdx.x * blockDim.x + threadIdx.x;
    size_t per = (size_t)s * s;
    size_t total = 4u * per;
    if (idx >= total) return;

    int l = (int)(idx / per);
    size_t p = idx - (size_t)l * per;
    int y = (int)(p / s);
    int x = (int)(p - (size_t)y * s);

    int half = s >> 1;
    int yy = (y + half) & (s - 1);   // s is a power of two
    int xx = (x + half) & (s - 1);

    float theta = ((float)M_PI * (float)l) / 4.0f;   // matches jnp.pi*arange(L)/L in f32
    float ct = cosf(theta), st = sinf(theta);

    float gy = (2.0f * (float)yy + 1.0f) / (float)s - 1.0f;
    float gx = (2.0f * (float)xx + 1.0f) / (float)s - 1.0f;
    float sx = ct * gx - st * gy;
    float sy = st * gx + ct * gy;
    float ix = ((sx + 1.0f) * (float)s - 1.0f) * 0.5f;
    float iy = ((sy + 1.0f) * (float)s - 1.0f) * 0.5f;

    float y0 = floorf(iy), x0 = floorf(ix);
    float wy = iy - y0,    wx = ix - x0;
    int y0i = (int)y0,     x0i = (int)x0;

    const float* Rm = R;
    const float* Rp = R + per;
    float m = 0.0f, ph = 0.0f;
    #pragma unroll
    for (int dy = 0; dy < 2; ++dy) {
        #pragma unroll
        for (int dx = 0; dx < 2; ++dx) {
            int yt = y0i + dy, xt = x0i + dx;
            float w = (dy ? wy : 1.0f - wy) * (dx ? wx : 1.0f - wx);
            if (yt >= 0 && yt < s && xt >= 0 && xt < s) {
                size_t o = (size_t)yt * s + xt;
                m  += Rm[o] * w;
                ph += Rp[o] * w;
            }
        }
    }

    float2 val;
    val.x = m * cosf(ph);
    val.y = m * sinf(ph);
    reinterpret_cast<float2*>(out)[outBaseComplex + idx] = val;
}

// ---------------------------------------------------------------------------
extern "C" void kernel_launch(void* const* d_in, const int* in_sizes, int n_in,
                              void* d_out, int out_size, void* d_ws, size_t ws_size,
                              hipStream_t stream) {
    const float* mags   = (const float*)d_in[0];
    const float* phases = (const float*)d_in[1];
    float* out = (float*)d_out;
    float* ws  = (float*)d_ws;

    static const int S[6] = {2048, 1024, 512, 256, 128, 64};

    // Workspace layout (floats):
    //   W_j matrices:   512 * sum(S) = 2,064,384
    //   T (2 images):   2 * 2048*512 = 2,097,152   (reused per scale)
    //   R (2 images):   2 * 2048*2048 = 8,388,608  (reused per scale)
    // Total ~50.2 MB.
    size_t woff[6]; size_t wtot = 0;
    for (int j = 0; j < 6; ++j) { woff[j] = wtot; wtot += (size_t)NSRC * S[j]; }
    float* Wbuf = ws;
    float* Tbuf = ws + wtot;
    float* Rbuf = Tbuf + (size_t)2 * 2048 * NSRC;

    // Stage 0: all weight matrices
    for (int j = 0; j < 6; ++j) {
        int s = S[j];
        int n = NSRC * s;
        weights_kernel<<<(n + 255) / 256, 256, 0, stream>>>(Wbuf + woff[j], s);
    }

    // Per scale: resize via WMMA GEMMs, then fused rotate/shift/complex
    unsigned long long outBase = 0;  // complex-element offset into d_out
    for (int j = 0; j < 6; ++j) {
        int s = S[j];

        int tiles1 = (s / 16) * (NSRC / 16);     // always multiple of 4
        dim3 g1((tiles1 + 3) / 4, 1, 2);
        gemm1_kernel<<<g1, 128, 0, stream>>>(mags, phases, Wbuf + woff[j], Tbuf, s);

        int tiles2 = (s / 16) * (s / 16);        // always multiple of 4
        dim3 g2((tiles2 + 3) / 4, 1, 2);
        gemm2_kernel<<<g2, 128, 0, stream>>>(Tbuf, Wbuf + woff[j], Rbuf, s);

        size_t total = 4ull * s * s;
        finalize_kernel<<<(unsigned)((total + 255) / 256), 256, 0, stream>>>(
            Rbuf, out, s, outBase);
        outBase += total;
    }
}